// SurfEval_20057497272879
// MI455X (gfx1250) — compile-verified
//
#include <hip/hip_runtime.h>

typedef float v2f __attribute__((ext_vector_type(2)));
typedef float v8f __attribute__((ext_vector_type(8)));

#define NCP  64
#define UPTS 1024
#define VPTS 1024
#define NB   4
#define VT_PER_WAVE 4   // v-tiles per wave in stage 2 (A-fragment reuse factor)

// ---------------------------------------------------------------------------
// Kernel 1: densify the banded B-spline basis tables.
// NuF[u][i] = Nu[u][l] at i = uspan[u]-3+l, else 0.  Same for NvF.
// ---------------------------------------------------------------------------
__global__ __launch_bounds__(256)
void build_dense_basis(const float* __restrict__ Nu,
                       const float* __restrict__ Nv,
                       const int*   __restrict__ uspan,
                       const int*   __restrict__ vspan,
                       float* __restrict__ NuF,
                       float* __restrict__ NvF) {
    int t = blockIdx.x * blockDim.x + threadIdx.x;
    if (t < UPTS) {
        float* row = NuF + t * NCP;
        #pragma unroll
        for (int i = 0; i < NCP; ++i) row[i] = 0.0f;
        int s = uspan[t] - 3;
        #pragma unroll
        for (int l = 0; l < 4; ++l) row[s + l] = Nu[t * 4 + l];
    } else if (t < 2 * UPTS) {
        int v = t - UPTS;
        float* row = NvF + v * NCP;
        #pragma unroll
        for (int i = 0; i < NCP; ++i) row[i] = 0.0f;
        int s = vspan[v] - 3;
        #pragma unroll
        for (int l = 0; l < 4; ++l) row[s + l] = Nv[v * 4 + l];
    }
}

// ---------------------------------------------------------------------------
// Kernel 2: T[b,d][u][j] = sum_i NuF[u][i] * ctrl[b][i][j][d]
// One wave32 computes one 16x16 (u x j) tile via 16x V_WMMA_F32_16X16X4_F32.
// grid = (u_tiles=64, j_tiles=4, b*4+d=16), block = 32.
// ---------------------------------------------------------------------------
__global__ __launch_bounds__(32)
void surf_stage1(const float* __restrict__ NuF,
                 const float* __restrict__ ctrl,   // [B][NCP][NCP][4]
                 float* __restrict__ T) {          // [B*4][UPTS][NCP]
    const int lane = threadIdx.x;           // 0..31
    const int ut   = blockIdx.x;            // u tile
    const int jt   = blockIdx.y;            // j tile
    const int bd   = blockIdx.z;            // b*4 + d
    const int b    = bd >> 2;
    const int d    = bd & 3;

    const int half = lane >> 4;             // 0 or 1
    const int l15  = lane & 15;
    const int urow = ut * 16 + l15;         // A-matrix row held by this lane
    const int jcol = jt * 16 + l15;         // B-matrix col held by this lane

    v8f acc = {};
    const float* nurow = NuF + urow * NCP;

    #pragma unroll
    for (int k0 = 0; k0 < NCP; k0 += 4) {
        const int ka = k0 + half * 2;       // this lane's K pair
        // A fragment: NuF rows, 8-byte aligned pair load
        v2f a = *(const v2f*)(nurow + ka);
        // B fragment: ctrl[b][i=ka(+1)][jcol][d]
        v2f bf;
        bf.x = ctrl[((b * NCP + ka    ) * NCP + jcol) * 4 + d];
        bf.y = ctrl[((b * NCP + ka + 1) * NCP + jcol) * 4 + d];
        acc = __builtin_amdgcn_wmma_f32_16x16x4_f32(
                  false, a, false, bf, (short)0, acc, false, false);
    }

    // D layout: VGPR r holds M = half*8 + r, N = l15
    #pragma unroll
    for (int r = 0; r < 8; ++r) {
        const int u = ut * 16 + half * 8 + r;
        T[(bd * UPTS + u) * NCP + (jt * 16 + l15)] = acc[r];
    }
}

// ---------------------------------------------------------------------------
// Kernel 3: Sw[b][u][v][d] = sum_j T[b,d][u][j] * NvF[v][j]; out = xyz / w.
// One wave32 computes a 16x64 (u x v) strip: VT_PER_WAVE v-tiles x 4 dims
// = 16 accumulators.  Per K-step: 4 A-fragment loads (T rows, one per dim,
// reused across all v-tiles) + VT_PER_WAVE B-fragment loads feed 16 WMMAs
// -> 0.5 loads/wmma and 4x less T re-read traffic.  Perspective divide
// fused into the store.  grid = (16, 64, 4), block = 32 (one wave).
// ---------------------------------------------------------------------------
__global__ __launch_bounds__(32)
void surf_stage2(const float* __restrict__ T,     // [B*4][UPTS][NCP]
                 const float* __restrict__ NvF,   // [VPTS][NCP]
                 float* __restrict__ out) {       // [B][UPTS][VPTS][3]
    const int lane = threadIdx.x;
    const int vt0  = blockIdx.x * VT_PER_WAVE;
    const int ut   = blockIdx.y;
    const int b    = blockIdx.z;

    const int half = lane >> 4;
    const int l15  = lane & 15;
    const int urow = ut * 16 + l15;

    const float* t0 = T + ((b * 4 + 0) * UPTS + urow) * NCP;
    const float* t1 = T + ((b * 4 + 1) * UPTS + urow) * NCP;
    const float* t2 = T + ((b * 4 + 2) * UPTS + urow) * NCP;
    const float* t3 = T + ((b * 4 + 3) * UPTS + urow) * NCP;

    // B-fragment row pointers, one per v-tile
    const float* nvr[VT_PER_WAVE];
    #pragma unroll
    for (int iv = 0; iv < VT_PER_WAVE; ++iv)
        nvr[iv] = NvF + ((vt0 + iv) * 16 + l15) * NCP;

    v8f acc[VT_PER_WAVE][4];
    #pragma unroll
    for (int iv = 0; iv < VT_PER_WAVE; ++iv)
        #pragma unroll
        for (int d = 0; d < 4; ++d)
            acc[iv][d] = (v8f){};

    #pragma unroll
    for (int k0 = 0; k0 < NCP; k0 += 4) {
        const int ka = k0 + half * 2;
        // A fragments (T rows) — loaded once, reused for every v-tile
        v2f a0 = *(const v2f*)(t0 + ka);
        v2f a1 = *(const v2f*)(t1 + ka);
        v2f a2 = *(const v2f*)(t2 + ka);
        v2f a3 = *(const v2f*)(t3 + ka);
        #pragma unroll
        for (int iv = 0; iv < VT_PER_WAVE; ++iv) {
            // B fragment: B[k][n] = NvF[vcol][k]  (NvF^T)
            v2f bf = *(const v2f*)(nvr[iv] + ka);
            acc[iv][0] = __builtin_amdgcn_wmma_f32_16x16x4_f32(
                             false, a0, false, bf, (short)0, acc[iv][0], false, false);
            acc[iv][1] = __builtin_amdgcn_wmma_f32_16x16x4_f32(
                             false, a1, false, bf, (short)0, acc[iv][1], false, false);
            acc[iv][2] = __builtin_amdgcn_wmma_f32_16x16x4_f32(
                             false, a2, false, bf, (short)0, acc[iv][2], false, false);
            acc[iv][3] = __builtin_amdgcn_wmma_f32_16x16x4_f32(
                             false, a3, false, bf, (short)0, acc[iv][3], false, false);
        }
    }

    // Perspective divide + store: out[b][u][v][0..2] = Sw.xyz / Sw.w
    #pragma unroll
    for (int iv = 0; iv < VT_PER_WAVE; ++iv) {
        #pragma unroll
        for (int r = 0; r < 8; ++r) {
            const int u = ut * 16 + half * 8 + r;
            const int v = (vt0 + iv) * 16 + l15;
            const float w = acc[iv][3][r];
            const size_t o = ((size_t)(b * UPTS + u) * VPTS + v) * 3;
            out[o + 0] = acc[iv][0][r] / w;
            out[o + 1] = acc[iv][1][r] / w;
            out[o + 2] = acc[iv][2][r] / w;
        }
    }
}

// ---------------------------------------------------------------------------
// Launcher
// ---------------------------------------------------------------------------
extern "C" void kernel_launch(void* const* d_in, const int* in_sizes, int n_in,
                              void* d_out, int out_size, void* d_ws, size_t ws_size,
                              hipStream_t stream) {
    const float* ctrl  = (const float*)d_in[0];   // (4,64,64,4) f32
    const float* Nu    = (const float*)d_in[1];   // (1024,4) f32
    const float* Nv    = (const float*)d_in[2];   // (1024,4) f32
    const int*   uspan = (const int*)  d_in[3];   // (1024,) i32
    const int*   vspan = (const int*)  d_in[4];   // (1024,) i32
    float*       out   = (float*)d_out;           // (4,1024,1024,3) f32

    // Workspace layout: NuF | NvF | T   (256KB + 256KB + 4MB)
    float* NuF = (float*)d_ws;
    float* NvF = NuF + (size_t)UPTS * NCP;
    float* T   = NvF + (size_t)VPTS * NCP;

    build_dense_basis<<<(2 * UPTS + 255) / 256, 256, 0, stream>>>(
        Nu, Nv, uspan, vspan, NuF, NvF);

    surf_stage1<<<dim3(UPTS / 16, NCP / 16, NB * 4), 32, 0, stream>>>(
        NuF, ctrl, T);

    surf_stage2<<<dim3(VPTS / 16 / VT_PER_WAVE, UPTS / 16, NB), 32, 0, stream>>>(
        T, NvF, out);
}